// TD_BERT_with_GCN_73375221285290
// MI455X (gfx1250) — compile-verified
//
#include <hip/hip_runtime.h>

// Problem constants (match reference)
#define BB 64
#define SS 512
#define NN 128
#define DD 768
#define OO 3

typedef float v2f __attribute__((ext_vector_type(2)));
typedef float v8f __attribute__((ext_vector_type(8)));

// ---------------------------------------------------------------------------
// CDNA5 async copy: global -> LDS, 16B per lane, tracked by ASYNCcnt.
// lds_off is the raw LDS byte address (low 32 bits of the flat address of a
// __shared__ element == DS offset per the CDNA5 aperture mapping).
// ---------------------------------------------------------------------------
__device__ __forceinline__ void async_ld_b128(unsigned lds_off, const float* g)
{
    asm volatile("global_load_async_to_lds_b128 %0, %1, off"
                 :: "v"(lds_off), "v"(g)
                 : "memory");
}
__device__ __forceinline__ void wait_async0()
{
    asm volatile("s_wait_asynccnt 0x0" ::: "memory");
}
__device__ __forceinline__ unsigned lds_addr(const void* p)
{
    return (unsigned)(unsigned long long)p;   // low 32 bits = LDS offset
}

// ---------------------------------------------------------------------------
// Prep: ragged target max-pool over sentence_embed[l:r) -> tmax [B,D]
// ---------------------------------------------------------------------------
__global__ __launch_bounds__(256) void prep_tmax_kernel(
    const float* __restrict__ SE, const int* __restrict__ tspan,
    float* __restrict__ tmax)
{
    int b = blockIdx.x;
    int l = tspan[b * 2 + 0];
    int r = tspan[b * 2 + 1];
    for (int d = threadIdx.x; d < DD; d += 256) {
        float m = -__FLT_MAX__;
        for (int s = l; s < r; ++s)
            m = fmaxf(m, SE[((long)b * SS + s) * DD + d]);
        tmax[(long)b * DD + d] = m;
    }
}

// ---------------------------------------------------------------------------
// Prep: contiguous span sums (tran_indices): tmps[b,n,:] = sum SE[b, s0+1:s1+1, :]
// ---------------------------------------------------------------------------
__global__ __launch_bounds__(256) void prep_tmps_kernel(
    const float* __restrict__ SE, const int* __restrict__ spans,
    float* __restrict__ tmps)
{
    int b = blockIdx.x, n = blockIdx.y;
    int s0 = spans[((long)b * NN + n) * 2 + 0] + 1;
    int s1 = spans[((long)b * NN + n) * 2 + 1] + 1;
    for (int d = threadIdx.x; d < DD; d += 256) {
        float acc = 0.f;
        for (int s = s0; s < s1; ++s)
            acc += SE[((long)b * SS + s) * DD + d];
        tmps[((long)b * NN + n) * DD + d] = acc;
    }
}

// ---------------------------------------------------------------------------
// Prep: adj = min(dg+dg1, 1); rdenom = 1 / (rowsum(adj) + 1e-7)
// ---------------------------------------------------------------------------
__global__ __launch_bounds__(128) void prep_adj_kernel(
    const float* __restrict__ dg, const float* __restrict__ dg1,
    float* __restrict__ adj, float* __restrict__ rdenom)
{
    int b = blockIdx.x, n = blockIdx.y;
    int t = threadIdx.x;
    long idx = ((long)b * NN + n) * NN + t;
    float v = dg[idx] + dg1[idx];
    v = (v >= 1.f) ? 1.f : v;
    adj[idx] = v;
    __shared__ float red[128];
    red[t] = v;
    __syncthreads();
    for (int off = 64; off > 0; off >>= 1) {
        if (t < off) red[t] += red[t + off];
        __syncthreads();
    }
    if (t == 0) rdenom[(long)b * NN + n] = 1.f / (red[0] + 1e-7f);
}

// ---------------------------------------------------------------------------
// Core f32 WMMA GEMM:  C[M,N] = (relu?)(A[M,K] @ B[K,N]),  lda=K, ldb=ldc=N.
// Block tile 128x64, 8 wave32s in a 4(M)x2(N) grid; each wave computes a
// 32x32 region via a 2x2 grid of v_wmma_f32_16x16x4_f32 accumulators, so
// every A fragment feeds two WMMAs and every B fragment feeds two WMMAs
// (halves LDS fragment traffic per matrix op vs a 16x16 wave tile).
// K staged 32-wide via LDS with double-buffered async global->LDS copies
// (ASYNCcnt): chunk k+1 is in flight while chunk k feeds the WMMAs; one
// barrier per chunk.
//   A frag (16x4 f32): lane holds A[M=lane&15][K = 2*(lane>>4) + {0,1}]
//   B frag (4x16 f32): lane holds B[K = 2*(lane>>4) + {0,1}][N=lane&15]
//   C/D   (16x16 f32): elem j -> row j + 8*(lane>>4), col lane&15
// LDS pads: A stride 36, B stride 72 -> conflict-free fragment gathers and
// 16B-aligned async destinations (144 B / 288 B row pitches).
// ---------------------------------------------------------------------------
template <bool RELU>
__global__ __launch_bounds__(256) void gemm_f32_wmma_kernel(
    const float* __restrict__ A, const float* __restrict__ B,
    float* __restrict__ C, int M, int K, int N)
{
    __shared__ __align__(16) float As[2][128 * 36];
    __shared__ __align__(16) float Bs[2][32 * 72];

    const int tid  = threadIdx.x;
    const int lane = tid & 31;
    const int w    = tid >> 5;
    const int wm   = w >> 1;          // 0..3  -> M offset wm*32
    const int wn   = w & 1;           // 0..1  -> N offset wn*32
    const int m0   = blockIdx.y * 128;
    const int n0   = blockIdx.x * 64;
    const int lm   = lane & 15;
    const int kh   = lane >> 4;       // 0/1

    // Per-thread staging coordinates: 4 x b128 for A (128x32), 2 x b128 for B (32x64).
    int raf[4], caf[4];
    const float* gA[4];
    unsigned lA[4][2];
#pragma unroll
    for (int j = 0; j < 4; ++j) {
        int f = (tid + j * 256) * 4;
        raf[j] = f >> 5; caf[j] = f & 31;
        gA[j] = A + (long)(m0 + raf[j]) * K + caf[j];
        lA[j][0] = lds_addr(&As[0][raf[j] * 36 + caf[j]]);
        lA[j][1] = lds_addr(&As[1][raf[j] * 36 + caf[j]]);
    }
    const int f0 = tid * 4, f1 = (tid + 256) * 4;
    const int rb0 = f0 >> 6, cb0 = f0 & 63, rb1 = f1 >> 6, cb1 = f1 & 63;
    const float* gB0 = B + (long)rb0 * N + n0 + cb0;
    const float* gB1 = B + (long)rb1 * N + n0 + cb1;
    const unsigned lB0[2] = { lds_addr(&Bs[0][rb0 * 72 + cb0]), lds_addr(&Bs[1][rb0 * 72 + cb0]) };
    const unsigned lB1[2] = { lds_addr(&Bs[0][rb1 * 72 + cb1]), lds_addr(&Bs[1][rb1 * 72 + cb1]) };

    v8f acc00 = {}, acc01 = {}, acc10 = {}, acc11 = {};
    const int nk = K >> 5;

    // prologue: stage chunk 0 into buffer 0
#pragma unroll
    for (int j = 0; j < 4; ++j) async_ld_b128(lA[j][0], gA[j]);
    async_ld_b128(lB0[0], gB0);
    async_ld_b128(lB1[0], gB1);

    for (int kc = 0; kc < nk; ++kc) {
        wait_async0();          // this wave's slice of buf[kc&1] has landed
        __syncthreads();        // everyone's slice landed; prev compute done
        if (kc + 1 < nk) {      // prefetch next chunk into the other buffer
            int k0 = (kc + 1) << 5;
            int nb = (kc + 1) & 1;
#pragma unroll
            for (int j = 0; j < 4; ++j) async_ld_b128(lA[j][nb], gA[j] + k0);
            async_ld_b128(lB0[nb], gB0 + (long)k0 * N);
            async_ld_b128(lB1[nb], gB1 + (long)k0 * N);
        }
        const float* as = As[kc & 1];
        const float* bs = Bs[kc & 1];
#pragma unroll
        for (int kk = 0; kk < 8; ++kk) {
            int kb = kk * 4 + kh * 2;
            const float* ap0 = &as[(wm * 32 + lm) * 36 + kb];
            const float* ap1 = ap0 + 16 * 36;
            v2f a0; a0.x = ap0[0]; a0.y = ap0[1];
            v2f a1; a1.x = ap1[0]; a1.y = ap1[1];
            const float* bp = &bs[kb * 72 + wn * 32 + lm];
            v2f b0; b0.x = bp[0];  b0.y = bp[72];
            v2f b1; b1.x = bp[16]; b1.y = bp[72 + 16];
            acc00 = __builtin_amdgcn_wmma_f32_16x16x4_f32(
                false, a0, false, b0, (short)0, acc00, false, false);
            acc01 = __builtin_amdgcn_wmma_f32_16x16x4_f32(
                false, a0, false, b1, (short)0, acc01, false, false);
            acc10 = __builtin_amdgcn_wmma_f32_16x16x4_f32(
                false, a1, false, b0, (short)0, acc10, false, false);
            acc11 = __builtin_amdgcn_wmma_f32_16x16x4_f32(
                false, a1, false, b1, (short)0, acc11, false, false);
        }
    }

    const int r0 = m0 + wm * 32 + kh * 8;     // strip 0 rows
    const int r1 = r0 + 16;                   // strip 1 rows
    const int cc = n0 + wn * 32 + lm;
#pragma unroll
    for (int j = 0; j < 8; ++j) {
        float v00 = acc00[j], v01 = acc01[j], v10 = acc10[j], v11 = acc11[j];
        if (RELU) {
            v00 = fmaxf(v00, 0.f); v01 = fmaxf(v01, 0.f);
            v10 = fmaxf(v10, 0.f); v11 = fmaxf(v11, 0.f);
        }
        C[(long)(r0 + j) * N + cc]      = v00;
        C[(long)(r0 + j) * N + cc + 16] = v01;
        C[(long)(r1 + j) * N + cc]      = v10;
        C[(long)(r1 + j) * N + cc + 16] = v11;
    }
}

// ---------------------------------------------------------------------------
// Batched GCN combine:  X[b] = relu(adj[b] @ HW[b] * rdenom[b,:,None] + bias) + X[b]
// M=K=128 (nodes), N=768. Same 128x64 block tile (one block row covers all
// nodes), 32x32 wave tiles, async double-buffered staging. In-place residual
// update is safe: each element read-then-written by exactly one thread.
// ---------------------------------------------------------------------------
__global__ __launch_bounds__(256) void gcn_combine_wmma_kernel(
    const float* __restrict__ adj, const float* __restrict__ HW,
    float* __restrict__ X, const float* __restrict__ rdenom,
    const float* __restrict__ bias)
{
    const int b = blockIdx.z;
    const float* Ab = adj + (long)b * NN * NN;     // 128x128, lda=128
    const float* Bb = HW  + (long)b * NN * DD;     // 128x768, ldb=768
    float*       Xb = X   + (long)b * NN * DD;
    const float* rd = rdenom + (long)b * NN;

    __shared__ __align__(16) float As[2][128 * 36];
    __shared__ __align__(16) float Bs[2][32 * 72];

    const int tid  = threadIdx.x;
    const int lane = tid & 31;
    const int w    = tid >> 5;
    const int wm   = w >> 1;
    const int wn   = w & 1;
    const int n0   = blockIdx.x * 64;   // 0..704
    const int lm   = lane & 15;
    const int kh   = lane >> 4;

    int raf[4], caf[4];
    const float* gA[4];
    unsigned lA[4][2];
#pragma unroll
    for (int j = 0; j < 4; ++j) {
        int f = (tid + j * 256) * 4;
        raf[j] = f >> 5; caf[j] = f & 31;
        gA[j] = Ab + (long)raf[j] * NN + caf[j];
        lA[j][0] = lds_addr(&As[0][raf[j] * 36 + caf[j]]);
        lA[j][1] = lds_addr(&As[1][raf[j] * 36 + caf[j]]);
    }
    const int f0 = tid * 4, f1 = (tid + 256) * 4;
    const int rb0 = f0 >> 6, cb0 = f0 & 63, rb1 = f1 >> 6, cb1 = f1 & 63;
    const float* gB0 = Bb + (long)rb0 * DD + n0 + cb0;
    const float* gB1 = Bb + (long)rb1 * DD + n0 + cb1;
    const unsigned lB0[2] = { lds_addr(&Bs[0][rb0 * 72 + cb0]), lds_addr(&Bs[1][rb0 * 72 + cb0]) };
    const unsigned lB1[2] = { lds_addr(&Bs[0][rb1 * 72 + cb1]), lds_addr(&Bs[1][rb1 * 72 + cb1]) };

    v8f acc00 = {}, acc01 = {}, acc10 = {}, acc11 = {};
    const int nk = NN >> 5;             // 4 chunks

#pragma unroll
    for (int j = 0; j < 4; ++j) async_ld_b128(lA[j][0], gA[j]);
    async_ld_b128(lB0[0], gB0);
    async_ld_b128(lB1[0], gB1);

    for (int kc = 0; kc < nk; ++kc) {
        wait_async0();
        __syncthreads();
        if (kc + 1 < nk) {
            int k0 = (kc + 1) << 5;
            int nb = (kc + 1) & 1;
#pragma unroll
            for (int j = 0; j < 4; ++j) async_ld_b128(lA[j][nb], gA[j] + k0);
            async_ld_b128(lB0[nb], gB0 + (long)k0 * DD);
            async_ld_b128(lB1[nb], gB1 + (long)k0 * DD);
        }
        const float* as = As[kc & 1];
        const float* bs = Bs[kc & 1];
#pragma unroll
        for (int kk = 0; kk < 8; ++kk) {
            int kb = kk * 4 + kh * 2;
            const float* ap0 = &as[(wm * 32 + lm) * 36 + kb];
            const float* ap1 = ap0 + 16 * 36;
            v2f a0; a0.x = ap0[0]; a0.y = ap0[1];
            v2f a1; a1.x = ap1[0]; a1.y = ap1[1];
            const float* bp = &bs[kb * 72 + wn * 32 + lm];
            v2f b0; b0.x = bp[0];  b0.y = bp[72];
            v2f b1; b1.x = bp[16]; b1.y = bp[72 + 16];
            acc00 = __builtin_amdgcn_wmma_f32_16x16x4_f32(
                false, a0, false, b0, (short)0, acc00, false, false);
            acc01 = __builtin_amdgcn_wmma_f32_16x16x4_f32(
                false, a0, false, b1, (short)0, acc01, false, false);
            acc10 = __builtin_amdgcn_wmma_f32_16x16x4_f32(
                false, a1, false, b0, (short)0, acc10, false, false);
            acc11 = __builtin_amdgcn_wmma_f32_16x16x4_f32(
                false, a1, false, b1, (short)0, acc11, false, false);
        }
    }

    const int r0 = wm * 32 + kh * 8;
    const int r1 = r0 + 16;
    const int cc = n0 + wn * 32 + lm;
    const float bia0 = bias[cc];
    const float bia1 = bias[cc + 16];
#pragma unroll
    for (int j = 0; j < 8; ++j) {
        int ra = r0 + j, rb = r1 + j;
        float rda = rd[ra], rdb = rd[rb];
        long ia = (long)ra * DD + cc;
        long ib = (long)rb * DD + cc;
        float v00 = fmaxf(acc00[j] * rda + bia0, 0.f) + Xb[ia];
        float v01 = fmaxf(acc01[j] * rda + bia1, 0.f) + Xb[ia + 16];
        float v10 = fmaxf(acc10[j] * rdb + bia0, 0.f) + Xb[ib];
        float v11 = fmaxf(acc11[j] * rdb + bia1, 0.f) + Xb[ib + 16];
        Xb[ia]      = v00;
        Xb[ia + 16] = v01;
        Xb[ib]      = v10;
        Xb[ib + 16] = v11;
    }
}

// ---------------------------------------------------------------------------
// Head: gcn_target[b,:] = sum_{n in [gs,ge)} X[b,n,:];
// logits = tanh(concat(tmax, gcn_target) @ fc_W + fc_b)
// ---------------------------------------------------------------------------
__global__ __launch_bounds__(256) void head_kernel(
    const float* __restrict__ X, const float* __restrict__ tmax,
    const int* __restrict__ gspan, const float* __restrict__ fcW,
    const float* __restrict__ fcb, float* __restrict__ out)
{
    int b = blockIdx.x;
    int t = threadIdx.x;
    int gs = gspan[b * 2 + 0];
    int ge = gspan[b * 2 + 1];
    float p0 = 0.f, p1 = 0.f, p2 = 0.f;
    for (int d = t; d < DD; d += 256) {
        float g = 0.f;
        for (int n = gs; n < ge; ++n)
            g += X[((long)b * NN + n) * DD + d];
        float tm = tmax[(long)b * DD + d];
        p0 += tm * fcW[d * OO + 0] + g * fcW[(DD + d) * OO + 0];
        p1 += tm * fcW[d * OO + 1] + g * fcW[(DD + d) * OO + 1];
        p2 += tm * fcW[d * OO + 2] + g * fcW[(DD + d) * OO + 2];
    }
    __shared__ float r0[256], r1[256], r2[256];
    r0[t] = p0; r1[t] = p1; r2[t] = p2;
    __syncthreads();
    for (int off = 128; off > 0; off >>= 1) {
        if (t < off) { r0[t] += r0[t + off]; r1[t] += r1[t + off]; r2[t] += r2[t + off]; }
        __syncthreads();
    }
    if (t == 0) {
        out[b * OO + 0] = tanhf(r0[0] + fcb[0]);
        out[b * OO + 1] = tanhf(r1[0] + fcb[1]);
        out[b * OO + 2] = tanhf(r2[0] + fcb[2]);
    }
}

// ---------------------------------------------------------------------------
extern "C" void kernel_launch(void* const* d_in, const int* in_sizes, int n_in,
                              void* d_out, int out_size, void* d_ws, size_t ws_size,
                              hipStream_t stream)
{
    const float* SE    = (const float*)d_in[0];
    const float* dg    = (const float*)d_in[1];
    const float* dg1   = (const float*)d_in[2];
    const float* Wproj = (const float*)d_in[3];
    const float* Wg1   = (const float*)d_in[4];
    const float* bg1   = (const float*)d_in[5];
    const float* Wg2   = (const float*)d_in[6];
    const float* bg2   = (const float*)d_in[7];
    const float* Wg3   = (const float*)d_in[8];
    const float* bg3   = (const float*)d_in[9];
    const float* fcW   = (const float*)d_in[10];
    const float* fcb   = (const float*)d_in[11];
    const int*   tspan = (const int*)d_in[12];
    const int*   nspan = (const int*)d_in[13];
    const int*   gspan = (const int*)d_in[14];
    float* out = (float*)d_out;

    float* ws     = (float*)d_ws;
    float* tmax   = ws;                                   // B*D
    float* rdenom = tmax   + (size_t)BB * DD;             // B*N
    float* adj    = rdenom + (size_t)BB * NN;             // B*N*N
    float* tmps   = adj    + (size_t)BB * NN * NN;        // B*N*D
    float* X      = tmps   + (size_t)BB * NN * DD;        // B*N*D
    float* HW     = X      + (size_t)BB * NN * DD;        // B*N*D

    // prep
    prep_tmax_kernel<<<BB, 256, 0, stream>>>(SE, tspan, tmax);
    prep_tmps_kernel<<<dim3(BB, NN), 256, 0, stream>>>(SE, nspan, tmps);
    prep_adj_kernel<<<dim3(BB, NN), 128, 0, stream>>>(dg, dg1, adj, rdenom);

    const int M = BB * NN;                      // 8192
    dim3 ggrid(DD / 64, M / 128);               // 12 x 64
    dim3 cgrid(DD / 64, 1, BB);                 // 12 x 1 x 64

    // X0 = relu(tmps @ W_proj)
    gemm_f32_wmma_kernel<true><<<ggrid, 256, 0, stream>>>(tmps, Wproj, X, M, DD, DD);

    // three GCN layers: HW = X @ Wg;  X = relu(adj@HW / denom + b) + X
    gemm_f32_wmma_kernel<false><<<ggrid, 256, 0, stream>>>(X, Wg1, HW, M, DD, DD);
    gcn_combine_wmma_kernel<<<cgrid, 256, 0, stream>>>(adj, HW, X, rdenom, bg1);

    gemm_f32_wmma_kernel<false><<<ggrid, 256, 0, stream>>>(X, Wg2, HW, M, DD, DD);
    gcn_combine_wmma_kernel<<<cgrid, 256, 0, stream>>>(adj, HW, X, rdenom, bg2);

    gemm_f32_wmma_kernel<false><<<ggrid, 256, 0, stream>>>(X, Wg3, HW, M, DD, DD);
    gcn_combine_wmma_kernel<<<cgrid, 256, 0, stream>>>(adj, HW, X, rdenom, bg3);

    // head
    head_kernel<<<BB, 256, 0, stream>>>(X, tmax, gspan, fcW, fcb, out);
}